// MSFF_32160715113113
// MI455X (gfx1250) — compile-verified
//
#include <hip/hip_runtime.h>
#include <hip/hip_bf16.h>
#include <math.h>

typedef float v2f __attribute__((ext_vector_type(2)));
typedef float v8f __attribute__((ext_vector_type(8)));

#define CHN  64
#define HH   128
#define WW   128
#define EHH  64
#define EWW  64
#define QQ   (HH*WW)          // 16384
#define KNUM 9

// LDS row strides (floats)
#define SI  134   // 132-wide input rows (A-side)
#define SA  68    // 64-wide activation rows (A-side)
#define PRS 160   // B-side pair-row stride: 128 payload + 32 pad (≡32 mod 64 banks)

// ---------------------------------------------------------------------------
// WMMA: D = A(16x4 f32) * B(4x16 f32) + C, one wave. [ISA 7.12.2 layouts]
// A frag: VGPR j, lane l -> (M = l&15, K = j + 2*(l>>4))
// B frag: VGPR j, lane l -> (K = j + 2*(l>>4), N = l&15)
// C/D:    VGPR i, lane l -> (M = i + 8*(l>>4), N = l&15)
// B staged in LDS pair-interleaved: s_wp[(k>>1)*PRS + n*2 + (k&1)]  -> one b64/lane
// ---------------------------------------------------------------------------
__device__ __forceinline__ v8f wmma_f32_k4(v2f a, v2f b, v8f c) {
  return __builtin_amdgcn_wmma_f32_16x16x4_f32(false, a, false, b, (short)0, c,
                                               false, false);
}

// ===========================================================================
// Kernel 1: per-pixel feature build + 4-layer MLP  -> feat (B,64,H,W)
// ===========================================================================
__device__ __forceinline__ void mlp_layer_lds(const float* s_in_, int sin, int kdim,
                                              const float* s_wp_, const float* bias,
                                              float* s_out_, bool relu,
                                              int m0, int hi, int nn)
{
  v8f acc[4];
#pragma unroll
  for (int nt = 0; nt < 4; ++nt) {
    float bv = bias[nt * 16 + nn];
#pragma unroll
    for (int i = 0; i < 8; ++i) acc[nt][i] = bv;
  }
  for (int kk = 0; kk < kdim; kk += 4) {
    const float* ap = s_in_ + (m0 + nn) * sin + kk + 2 * hi;
    v2f a; a.x = ap[0]; a.y = ap[1];
    const float* bp = s_wp_ + ((kk >> 1) + hi) * PRS + nn * 2;
#pragma unroll
    for (int nt = 0; nt < 4; ++nt) {
      v2f bf = *(const v2f*)(bp + nt * 32);
      acc[nt] = wmma_f32_k4(a, bf, acc[nt]);
    }
  }
#pragma unroll
  for (int nt = 0; nt < 4; ++nt)
#pragma unroll
    for (int i = 0; i < 8; ++i) {
      float v = acc[nt][i];
      if (relu) v = fmaxf(v, 0.0f);
      s_out_[(m0 + i + 8 * hi) * SA + nt * 16 + nn] = v;
    }
}

__launch_bounds__(256)
__global__ void msff_mlp_kernel(const float* __restrict__ ev, const float* __restrict__ im,
                                const float* __restrict__ w0, const float* __restrict__ b0,
                                const float* __restrict__ w1, const float* __restrict__ b1,
                                const float* __restrict__ w2, const float* __restrict__ b2,
                                const float* __restrict__ w3, const float* __restrict__ b3,
                                float* __restrict__ feat)
{
  __shared__ float s_in[128 * SI];   // input tile; later reused as act buffer B
  __shared__ float s_a[128 * SA];    // act buffer A
  __shared__ float s_wp[66 * PRS];   // current layer weights, pair-interleaved

  const int tid  = threadIdx.x;
  const int wid  = tid >> 5;
  const int lane = tid & 31;
  const int hi   = lane >> 4;
  const int nn   = lane & 15;
  const int m0   = wid * 16;

  const int r0 = blockIdx.x * 128;   // 128 pixel-rows per block, 512 blocks
  const int b  = r0 / QQ;
  const int q0 = r0 % QQ;

  // ---- build inp tile: [ev(64) | im(64) | rel_y rel_x | 1 1] ----
  for (int idx = tid; idx < 128 * 132; idx += 256) {
    int row = idx / 132;
    int col = idx - row * 132;
    int q = q0 + row;
    int h = q >> 7;
    int w = q & 127;
    float v;
    if (col < 64)        v = ev[((b * CHN + col) * EHH + (h >> 1)) * EWW + (w >> 1)];
    else if (col < 128)  v = im[((b * CHN + (col - 64)) * HH + h) * WW + w];
    else if (col == 128) v = (h & 1) ? 0.5f : -0.5f;   // rel_y
    else if (col == 129) v = (w & 1) ? 0.5f : -0.5f;   // rel_x
    else                 v = 1.0f;                     // rel_cell
    s_in[row * SI + col] = v;
  }
  for (int idx = tid; idx < 132 * 64; idx += 256) {
    int k = idx >> 6, n = idx & 63;
    s_wp[(k >> 1) * PRS + n * 2 + (k & 1)] = w0[idx];
  }
  // prefetch later-layer weights into cache while layer 0 runs
  __builtin_prefetch(w1 + tid * 16, 0, 3);
  __builtin_prefetch(w2 + tid * 16, 0, 3);
  __builtin_prefetch(w3 + tid * 16, 0, 3);
  __syncthreads();

  // layer 0: 132 -> 64, ReLU  (s_in -> s_a)
  mlp_layer_lds(s_in, SI, 132, s_wp, b0, s_a, true, m0, hi, nn);
  __syncthreads();
  for (int idx = tid; idx < 64 * 64; idx += 256) {
    int k = idx >> 6, n = idx & 63;
    s_wp[(k >> 1) * PRS + n * 2 + (k & 1)] = w1[idx];
  }
  __syncthreads();

  // layer 1: 64 -> 64, ReLU  (s_a -> s_in[reused])
  mlp_layer_lds(s_a, SA, 64, s_wp, b1, s_in, true, m0, hi, nn);
  __syncthreads();
  for (int idx = tid; idx < 64 * 64; idx += 256) {
    int k = idx >> 6, n = idx & 63;
    s_wp[(k >> 1) * PRS + n * 2 + (k & 1)] = w2[idx];
  }
  __syncthreads();

  // layer 2: 64 -> 64, ReLU  (s_in -> s_a)
  mlp_layer_lds(s_in, SA, 64, s_wp, b2, s_a, true, m0, hi, nn);
  __syncthreads();
  for (int idx = tid; idx < 64 * 64; idx += 256) {
    int k = idx >> 6, n = idx & 63;
    s_wp[(k >> 1) * PRS + n * 2 + (k & 1)] = w3[idx];
  }
  __syncthreads();

  // layer 3: 64 -> 64, no ReLU, store NCHW to global
  {
    v8f acc[4];
#pragma unroll
    for (int nt = 0; nt < 4; ++nt) {
      float bv = b3[nt * 16 + nn];
#pragma unroll
      for (int i = 0; i < 8; ++i) acc[nt][i] = bv;
    }
    for (int kk = 0; kk < 64; kk += 4) {
      const float* ap = s_a + (m0 + nn) * SA + kk + 2 * hi;
      v2f a; a.x = ap[0]; a.y = ap[1];
      const float* bp = s_wp + ((kk >> 1) + hi) * PRS + nn * 2;
#pragma unroll
      for (int nt = 0; nt < 4; ++nt) {
        v2f bf = *(const v2f*)(bp + nt * 32);
        acc[nt] = wmma_f32_k4(a, bf, acc[nt]);
      }
    }
#pragma unroll
    for (int nt = 0; nt < 4; ++nt) {
      int n = nt * 16 + nn;
#pragma unroll
      for (int i = 0; i < 8; ++i) {
        int q = q0 + m0 + i + 8 * hi;
        int h = q >> 7, w = q & 127;
        feat[((b * CHN + n) * HH + h) * WW + w] = acc[nt][i];
      }
    }
  }
}

// ===========================================================================
// Kernel 2: 3x3 conv 64 -> 27 (offsets raw, mask sigmoided) -> oo (B,27,H,W)
// 128 threads, 2 pixels per thread (rows 2*ty, 2*ty+1) for FMA density.
// ===========================================================================
__launch_bounds__(128)
__global__ void msff_offconv_kernel(const float* __restrict__ feat,
                                    const float* __restrict__ offw,
                                    const float* __restrict__ offb,
                                    float* __restrict__ oo)
{
  __shared__ float s_ft[64 * 324];   // 64 ch x 18x18 halo tile
  __shared__ float s_wo[27 * 576];   // all conv weights

  const int tid = threadIdx.x;
  const int blk = blockIdx.x;        // 4 * 64 tiles
  const int b   = blk >> 6;
  const int t   = blk & 63;
  const int th0 = (t >> 3) * 16;
  const int tw0 = (t & 7) * 16;

  for (int idx = tid; idx < 64 * 324; idx += 128) {
    int c = idx / 324;
    int p = idx - c * 324;
    int y = th0 + p / 18 - 1;
    int x = tw0 + (p % 18) - 1;
    float v = 0.0f;
    if (y >= 0 && y < HH && x >= 0 && x < WW)
      v = feat[((b * CHN + c) * HH + y) * WW + x];
    s_ft[idx] = v;
  }
  for (int idx = tid; idx < 27 * 576; idx += 128) s_wo[idx] = offw[idx];
  __syncthreads();

  const int ty = tid >> 4;           // 0..7  -> pixel rows 2*ty, 2*ty+1
  const int tx = tid & 15;
  float acc0[27], acc1[27];
#pragma unroll
  for (int oc = 0; oc < 27; ++oc) { float bv = offb[oc]; acc0[oc] = bv; acc1[oc] = bv; }

  for (int ic = 0; ic < 64; ++ic) {
    const float* fts = s_ft + ic * 324;
#pragma unroll
    for (int kh = 0; kh < 3; ++kh)
#pragma unroll
      for (int kw = 0; kw < 3; ++kw) {
        float v0 = fts[(2 * ty + kh) * 18 + (tx + kw)];
        float v1 = fts[(2 * ty + kh + 1) * 18 + (tx + kw)];
        int j = ic * 9 + kh * 3 + kw;
#pragma unroll
        for (int oc = 0; oc < 27; ++oc) {
          float wv = s_wo[oc * 576 + j];
          acc0[oc] += v0 * wv;
          acc1[oc] += v1 * wv;
        }
      }
  }
  const int h0 = th0 + 2 * ty, w = tw0 + tx;
#pragma unroll
  for (int oc = 0; oc < 27; ++oc) {
    float v0 = acc0[oc], v1 = acc1[oc];
    if (oc >= 18) {                               // mask channels pre-sigmoid
      v0 = 1.0f / (1.0f + expf(-v0));
      v1 = 1.0f / (1.0f + expf(-v1));
    }
    float* op = oo + ((b * 27 + oc) * HH + h0) * WW + w;
    op[0]  = v0;
    op[WW] = v1;
  }
}

// ===========================================================================
// Kernel 3: deformable bilinear sampling + (c,k)-contraction GEMM via WMMA
// ===========================================================================
__launch_bounds__(256)
__global__ void msff_dcn_kernel(const float* __restrict__ feat,
                                const float* __restrict__ oo,
                                const float* __restrict__ dcnw,
                                const float* __restrict__ dcnb,
                                float* __restrict__ out)
{
  __shared__ float s_chunk[128 * SA];  // 128 px x 64 ch sampled values (one tap)
  __shared__ float s_wp[32 * PRS];     // dcn_w slice for this tap, pair-interleaved
  __shared__ int   s_iy0[128], s_iy1[128], s_ix0[128], s_ix1[128];
  __shared__ float s_w00[128], s_w01[128], s_w10[128], s_w11[128];

  const int tid  = threadIdx.x;
  const int wid  = tid >> 5;
  const int lane = tid & 31;
  const int hi   = lane >> 4;
  const int nn   = lane & 15;
  const int m0   = wid * 16;

  const int r0 = blockIdx.x * 128;
  const int b  = r0 / QQ;
  const int q0 = r0 % QQ;
  const float* fb = feat + b * CHN * HH * WW;

  v8f acc[4];
#pragma unroll
  for (int nt = 0; nt < 4; ++nt) {
    float bv = dcnb[nt * 16 + nn];
#pragma unroll
    for (int i = 0; i < 8; ++i) acc[nt][i] = bv;
  }

  for (int k = 0; k < KNUM; ++k) {
    const int ky = k / 3 - 1;
    const int kx = k - (k / 3) * 3 - 1;

    if (tid < 128) {               // per-row bilinear metadata
      int row = tid;
      int q = q0 + row;
      int h = q >> 7, w = q & 127;
      const int cs = HH * WW;
      int base = (b * 27 * HH + h) * WW + w;
      float offy = oo[base + (2 * k) * cs];
      float offx = oo[base + (2 * k + 1) * cs];
      float mk   = oo[base + (18 + k) * cs];
      float py = (float)(h + ky) + offy;
      float px = (float)(w + kx) + offx;
      float y0 = floorf(py), x0 = floorf(px);
      float ly = py - y0,    lx = px - x0;
      float y1 = y0 + 1.0f,  x1 = x0 + 1.0f;
      float vy0 = (y0 >= 0.0f && y0 <= 127.0f) ? 1.0f : 0.0f;
      float vy1 = (y1 >= 0.0f && y1 <= 127.0f) ? 1.0f : 0.0f;
      float vx0 = (x0 >= 0.0f && x0 <= 127.0f) ? 1.0f : 0.0f;
      float vx1 = (x1 >= 0.0f && x1 <= 127.0f) ? 1.0f : 0.0f;
      s_iy0[row] = (int)fminf(fmaxf(y0, 0.0f), 127.0f);
      s_iy1[row] = (int)fminf(fmaxf(y1, 0.0f), 127.0f);
      s_ix0[row] = (int)fminf(fmaxf(x0, 0.0f), 127.0f);
      s_ix1[row] = (int)fminf(fmaxf(x1, 0.0f), 127.0f);
      s_w00[row] = (1.0f - ly) * (1.0f - lx) * mk * vy0 * vx0;
      s_w01[row] = (1.0f - ly) * lx * mk * vy0 * vx1;
      s_w10[row] = ly * (1.0f - lx) * mk * vy1 * vx0;
      s_w11[row] = ly * lx * mk * vy1 * vx1;
    }
    for (int idx = tid; idx < 4096; idx += 256) {   // stage B slice, pair layout
      int ic = idx >> 6, oc = idx & 63;
      s_wp[(ic >> 1) * PRS + oc * 2 + (ic & 1)] = dcnw[(oc * 64 + ic) * 9 + k];
    }
    if (k + 1 < KNUM)                               // prefetch next tap's slice
      __builtin_prefetch(dcnw + (tid * 16) * 9 + (k + 1), 0, 3);
    __syncthreads();

    {                              // build sampled chunk: 2 threads / row
      int row  = tid >> 1;
      int half = tid & 1;
      int a00 = s_iy0[row] * WW + s_ix0[row];
      int a01 = s_iy0[row] * WW + s_ix1[row];
      int a10 = s_iy1[row] * WW + s_ix0[row];
      int a11 = s_iy1[row] * WW + s_ix1[row];
      float c00 = s_w00[row], c01 = s_w01[row], c10 = s_w10[row], c11 = s_w11[row];
#pragma unroll 4
      for (int i = 0; i < 32; ++i) {
        int c = half * 32 + i;
        const float* fc = fb + c * (HH * WW);
        s_chunk[row * SA + c] =
            c00 * fc[a00] + c01 * fc[a01] + c10 * fc[a10] + c11 * fc[a11];
      }
    }
    __syncthreads();

    for (int kk = 0; kk < 64; kk += 4) {   // 16 k-steps x 4 n-tiles WMMA
      const float* ap = s_chunk + (m0 + nn) * SA + kk + 2 * hi;
      v2f a; a.x = ap[0]; a.y = ap[1];
      const float* bp = s_wp + ((kk >> 1) + hi) * PRS + nn * 2;
#pragma unroll
      for (int nt = 0; nt < 4; ++nt) {
        v2f bf = *(const v2f*)(bp + nt * 32);
        acc[nt] = wmma_f32_k4(a, bf, acc[nt]);
      }
    }
    __syncthreads();
  }

#pragma unroll
  for (int nt = 0; nt < 4; ++nt) {
    int n = nt * 16 + nn;
#pragma unroll
    for (int i = 0; i < 8; ++i) {
      int q = q0 + m0 + i + 8 * hi;
      int h = q >> 7, w = q & 127;
      out[((b * CHN + n) * HH + h) * WW + w] = acc[nt][i];
    }
  }
}

// ===========================================================================
extern "C" void kernel_launch(void* const* d_in, const int* in_sizes, int n_in,
                              void* d_out, int out_size, void* d_ws, size_t ws_size,
                              hipStream_t stream) {
  (void)in_sizes; (void)n_in; (void)out_size; (void)ws_size;
  const float* ev   = (const float*)d_in[0];
  const float* im   = (const float*)d_in[1];
  const float* w0   = (const float*)d_in[2];
  const float* b0   = (const float*)d_in[3];
  const float* w1   = (const float*)d_in[4];
  const float* b1   = (const float*)d_in[5];
  const float* w2   = (const float*)d_in[6];
  const float* b2   = (const float*)d_in[7];
  const float* w3   = (const float*)d_in[8];
  const float* b3   = (const float*)d_in[9];
  const float* offw = (const float*)d_in[10];
  const float* offb = (const float*)d_in[11];
  const float* dcnw = (const float*)d_in[12];
  const float* dcnb = (const float*)d_in[13];

  float* feat = (float*)d_ws;                        // 4*64*128*128 floats
  float* oo   = feat + 4 * CHN * HH * WW;            // 4*27*128*128 floats
  float* outp = (float*)d_out;

  msff_mlp_kernel<<<512, 256, 0, stream>>>(ev, im, w0, b0, w1, b1, w2, b2,
                                           w3, b3, feat);
  msff_offconv_kernel<<<256, 128, 0, stream>>>(feat, offw, offb, oo);
  msff_dcn_kernel<<<512, 256, 0, stream>>>(feat, oo, dcnw, dcnb, outp);
}